// SparseDropout_5171140624758
// MI455X (gfx1250) — compile-verified
//
#include <hip/hip_runtime.h>
#include <stdint.h>

// SparseDropout (dense-equivalent): out[i] = keep_mask[i] ? values[i] : 0
// Inputs (setup_inputs order): d_in[0]=indices int64 (2*NNZ)  [UNUSED — never read!]
//                              d_in[1]=values  float32 (NNZ)
//                              d_in[2]=keep_mask bool/1B (NNZ)
// Output: float32 (NNZ)
//
// Pure HBM-streaming kernel: 9 B/element, ~180 MB total -> ~7.7 us floor @ 23.3 TB/s.
// Uses the CDNA5 async global->LDS path (ASYNCcnt) with double buffering + NT stores.

typedef float v4f __attribute__((ext_vector_type(4)));

#define TILE_THREADS 256   // vec4s per tile == block size
#define GRID_BLOCKS  4096

// ---- CDNA5 async global->LDS (CDNA5 ISA), GV addressing ----
__device__ __forceinline__ void async_ld_b128(uint32_t lds_off, const void* gaddr) {
  asm volatile("global_load_async_to_lds_b128 %0, %1, off"
               :
               : "v"(lds_off), "v"((uint64_t)(uintptr_t)gaddr)
               : "memory");
}
__device__ __forceinline__ void async_ld_b32(uint32_t lds_off, const void* gaddr) {
  asm volatile("global_load_async_to_lds_b32 %0, %1, off"
               :
               : "v"(lds_off), "v"((uint64_t)(uintptr_t)gaddr)
               : "memory");
}
// Wait on this wave's ASYNCcnt (split-counter model on gfx1250)
__device__ __forceinline__ void wait_async_le2() {
  asm volatile("s_wait_asynccnt 0x2" ::: "memory");
}
__device__ __forceinline__ void wait_async_0() {
  asm volatile("s_wait_asynccnt 0x0" ::: "memory");
}

__device__ __forceinline__ v4f apply_mask(v4f v, uint32_t m) {
  v4f r;
  r.x = (m & 0x000000ffu) ? v.x : 0.0f;
  r.y = (m & 0x0000ff00u) ? v.y : 0.0f;
  r.z = (m & 0x00ff0000u) ? v.z : 0.0f;
  r.w = (m & 0xff000000u) ? v.w : 0.0f;
  return r;
}

__global__ void __launch_bounds__(TILE_THREADS)
sparse_dropout_async_kernel(const float* __restrict__ vals,
                            const unsigned char* __restrict__ msk,
                            float* __restrict__ out,
                            long long n) {
  // Double-buffered staging: 2 * (4 KB values + 1 KB mask) = 10 KB LDS / block.
  __shared__ v4f      s_val[2][TILE_THREADS];
  __shared__ uint32_t s_msk[2][TILE_THREADS];

  const int       tid   = threadIdx.x;
  const long long nvec  = n >> 2;                   // # of vec4 elements
  const long long nfull = nvec / TILE_THREADS;      // # of full tiles
  const long long gstride = (long long)gridDim.x;

  // Issue the 2 async staging loads (b128 values + b32 packed mask) for one tile.
  auto issue = [&](long long t, int buf) {
    const long long v = t * TILE_THREADS + tid;     // vec4 index for this lane
    async_ld_b128((uint32_t)(uintptr_t)&s_val[buf][tid], vals + v * 4);
    async_ld_b32 ((uint32_t)(uintptr_t)&s_msk[buf][tid], msk  + v * 4);
  };

  // ---- double-buffered async pipeline over full tiles (grid-stride) ----
  long long t   = (long long)blockIdx.x;
  int       buf = 0;
  if (t < nfull) issue(t, buf);                     // prologue: ASYNCcnt = 2
  while (t < nfull) {
    const long long nt = t + gstride;
    const bool has_next = (nt < nfull);
    if (has_next) {
      issue(nt, buf ^ 1);                           // ASYNCcnt = 4
      wait_async_le2();                             // current tile's 2 ops done
    } else {
      wait_async_0();
    }
    // Each lane consumes only its own staged bytes -> no barrier needed.
    const v4f      v = s_val[buf][tid];             // ds_load_b128
    const uint32_t m = s_msk[buf][tid];             // ds_load_b32
    const v4f      r = apply_mask(v, m);
    __builtin_nontemporal_store(r, (v4f*)(out + (t * TILE_THREADS + tid) * 4));
    t   = nt;
    buf ^= 1;
  }

  // ---- partial vec4 tile (direct NT path), handled by one designated block ----
  const long long rem_base = nfull * TILE_THREADS;
  if (rem_base < nvec &&
      (long long)blockIdx.x == (gstride ? (nfull % gstride) : 0)) {
    const long long v = rem_base + tid;
    if (v < nvec) {
      const v4f      val = __builtin_nontemporal_load((const v4f*)(vals + v * 4));
      const uint32_t m   = *(const uint32_t*)(msk + v * 4);
      __builtin_nontemporal_store(apply_mask(val, m), (v4f*)(out + v * 4));
    }
  }

  // ---- scalar tail (n % 4), block 0 only ----
  if (blockIdx.x == 0) {
    for (long long i = (nvec << 2) + tid; i < n; i += blockDim.x) {
      out[i] = msk[i] ? vals[i] : 0.0f;
    }
  }
}

extern "C" void kernel_launch(void* const* d_in, const int* in_sizes, int n_in,
                              void* d_out, int out_size, void* d_ws, size_t ws_size,
                              hipStream_t stream) {
  // d_in[0] (indices, int64, 2*NNZ = 320 MB) is intentionally never touched:
  // the dense-equivalent reference only masks `values`.
  const float*         vals = (const float*)d_in[1];
  const unsigned char* msk  = (const unsigned char*)d_in[2];
  float*               out  = (float*)d_out;
  const long long      n    = (long long)out_size;

  long long tiles = ((n >> 2) + TILE_THREADS - 1) / TILE_THREADS;
  int grid = GRID_BLOCKS;
  if (tiles > 0 && tiles < grid) grid = (int)tiles;
  if (grid < 1) grid = 1;

  sparse_dropout_async_kernel<<<grid, TILE_THREADS, 0, stream>>>(vals, msk, out, n);
}